// AdaptiveContactZones_11218454577887
// MI455X (gfx1250) — compile-verified
//
#include <hip/hip_runtime.h>
#include <hip/hip_bf16.h>
#include <math.h>

// ---------------------------------------------------------------------------
// AdaptiveContactZones for MI455X (gfx1250, wave32)
//
// d2(m,n) = |h|^2 + |o|^2 - 2 h.o computed as ONE V_WMMA_F32_16X16X4_F32 per
// 16x16 tile:
//   A[m][k] = [-2hx, -2hy, -2hz, |h|^2]   (16x4)
//   B[k][n] = [ox, oy, oz, 1]^T           (4x16)
//   C[m][n] = |o|^2 (broadcast per column: every VGPR of a lane is one column)
// so D = A*B + C = d2 straight off the matrix pipe. Running min kept in the
// D accumulator layout via raw v_min_num_f32. Loop control is scalarized
// (readfirstlane -> SALU loop, EXEC all-ones by construction) and unrolled
// x4 with immediate-offset loads for in-wave memory-level parallelism.
// ---------------------------------------------------------------------------

typedef __attribute__((ext_vector_type(2))) float v2f;
typedef __attribute__((ext_vector_type(8))) float v8f;

#define BATCH      32
#define NH         778
#define NV         8192
#define NTILES     (NV / 16)         // 512
#define HTILES     ((NH + 15) / 16)  // 49
#define WAVES      4
#define UNROLL     4
#define K_CONTACT  50
#define PROX       0.015f
#define BIGF       3.4e38f

__device__ __forceinline__ float fmin_raw(float a, float b) {
    float r;
    asm("v_min_num_f32 %0, %1, %2" : "=v"(r) : "v"(a), "v"(b));
    return r;
}

__global__ __launch_bounds__(WAVES * 32)
void acz_min_dist_kernel(const float* __restrict__ hand,
                         const float* __restrict__ obj,
                         float* __restrict__ out_min,
                         float* __restrict__ out_mask)
{
    const int b    = blockIdx.x;          // batch
    const int ht   = blockIdx.y;          // hand tile (0..48)
    const int lane = threadIdx.x & 31;
    const int half = lane >> 4;           // 0: lanes 0-15, 1: lanes 16-31
    const int idx  = lane & 15;
    // wave id as an SGPR so loop control is pure SALU (EXEC untouched)
    const int wv   = __builtin_amdgcn_readfirstlane((int)(threadIdx.x >> 5));

    // ---- load this lane's hand vertex (row M = idx of the tile) ----
    int m  = ht * 16 + idx;
    int mc = (m < NH) ? m : (NH - 1);     // clamp (padded rows not written)
    const float* hv = hand + ((size_t)b * NH + mc) * 3;
    const float hx = hv[0], hy = hv[1], hz = hv[2];
    const float h2 = hx * hx + hy * hy + hz * hz;

    // A-matrix (16x4 f32): lanes 0-15 hold K=0 (v0) / K=1 (v1),
    //                      lanes 16-31 hold K=2 (v0) / K=3 (v1).
    v2f a;
    a.x = half ? (-2.0f * hz) : (-2.0f * hx);
    a.y = half ? h2           : (-2.0f * hy);

    float mn[8];
#pragma unroll
    for (int r = 0; r < 8; ++r) mn[r] = BIGF;

    const float* ob = obj + (size_t)b * NV * 3;

    // Per-lane stream base: this wave's tiles are wv, wv+WAVES, ...
    const float* p = ob + (size_t)(wv * 16 + idx) * 3;
    const int STEP = WAVES * 16 * 3;      // floats between consecutive tiles

    // Warm caches for the second unrolled iteration (global_prefetch_b8);
    // the first iteration's tiles are loaded immediately below anyway.
    __builtin_prefetch(p + (size_t)UNROLL * STEP, 0, 3);
    __builtin_prefetch(p + (size_t)(UNROLL + 2) * STEP, 0, 3);

    // One 16x16 d2 tile: D = A*B + C, then running min.
    auto tile = [&](float ox, float oy, float oz) {
        const float o2 = ox * ox + oy * oy + oz * oz;
        v2f bb;
        bb.x = half ? oz : ox;
        bb.y = half ? 1.0f : oy;
        v8f c;
#pragma unroll
        for (int r = 0; r < 8; ++r) c[r] = o2;
        c = __builtin_amdgcn_wmma_f32_16x16x4_f32(
                false, a, false, bb, (short)0, c, false, false);
#pragma unroll
        for (int r = 0; r < 8; ++r) mn[r] = fmin_raw(mn[r], c[r]);
    };

    // Scalar loop: 512/(4 waves * 4 unroll) = 32 iterations, no remainder.
    for (int i = 0; i < NTILES / (WAVES * UNROLL); ++i) {
        // four independent b96 loads issue before the first wait (MLP=4)
        const float x0 = p[0 * STEP + 0], y0 = p[0 * STEP + 1], z0 = p[0 * STEP + 2];
        const float x1 = p[1 * STEP + 0], y1 = p[1 * STEP + 1], z1 = p[1 * STEP + 2];
        const float x2 = p[2 * STEP + 0], y2 = p[2 * STEP + 1], z2 = p[2 * STEP + 2];
        const float x3 = p[3 * STEP + 0], y3 = p[3 * STEP + 1], z3 = p[3 * STEP + 2];

        tile(x0, y0, z0);
        tile(x1, y1, z1);
        tile(x2, y2, z2);
        tile(x3, y3, z3);

        p += (size_t)UNROLL * STEP;
    }

    // ---- reduce over the 16 columns held in each half-wave ----
#pragma unroll
    for (int r = 0; r < 8; ++r) {
        float v = mn[r];
        v = fmin_raw(v, __shfl_xor(v, 1, 16));
        v = fmin_raw(v, __shfl_xor(v, 2, 16));
        v = fmin_raw(v, __shfl_xor(v, 4, 16));
        v = fmin_raw(v, __shfl_xor(v, 8, 16));
        mn[r] = v;  // row (half*8 + r) min, replicated in all lanes of half
    }

    __shared__ float smin[WAVES][16];
    if (idx == 0) {
#pragma unroll
        for (int r = 0; r < 8; ++r) smin[wv][half * 8 + r] = mn[r];
    }
    __syncthreads();

    // ---- combine waves, finalize sqrt + mask ----
    if (threadIdx.x < 16) {
        float v = smin[0][threadIdx.x];
#pragma unroll
        for (int w = 1; w < WAVES; ++w) v = fmin_raw(v, smin[w][threadIdx.x]);
        const int mo = ht * 16 + (int)threadIdx.x;
        if (mo < NH) {
            const float d = sqrtf(fmaxf(v, 0.0f));
            out_min[b * NH + mo]  = d;
            out_mask[b * NH + mo] = (d < PROX) ? 1.0f : 0.0f;
        }
    }
}

// ---------------------------------------------------------------------------
// Top-K (K=50) smallest per batch. One block per batch; 50 sequential
// argmin passes with index tie-break (first index wins, matching top_k on
// -min_dists: ascending distances). Work is trivial (32 * 50 * 778).
// ---------------------------------------------------------------------------
__global__ __launch_bounds__(256)
void acz_topk_kernel(const float* __restrict__ min_d,
                     int* __restrict__ out_idx,
                     float* __restrict__ out_dist)
{
    const int b   = blockIdx.x;
    const int tid = threadIdx.x;

    __shared__ float svals[NH];
    __shared__ float rv[256];
    __shared__ int   ri[256];

    for (int i = tid; i < NH; i += 256) svals[i] = min_d[b * NH + i];
    __syncthreads();

    for (int k = 0; k < K_CONTACT; ++k) {
        float best = BIGF;
        int   bi   = NH;
        for (int i = tid; i < NH; i += 256) {
            const float v = svals[i];
            if (v < best || (v == best && i < bi)) { best = v; bi = i; }
        }
        rv[tid] = best;
        ri[tid] = bi;
        __syncthreads();

        for (int s = 128; s > 0; s >>= 1) {
            if (tid < s) {
                const float v2 = rv[tid + s];
                const int   i2 = ri[tid + s];
                if (v2 < rv[tid] || (v2 == rv[tid] && i2 < ri[tid])) {
                    rv[tid] = v2;
                    ri[tid] = i2;
                }
            }
            __syncthreads();
        }

        if (tid == 0) {
            out_idx[b * K_CONTACT + k]  = ri[0];
            out_dist[b * K_CONTACT + k] = rv[0];
            svals[ri[0]] = BIGF;   // exclude from subsequent passes
        }
        __syncthreads();
    }
}

// ---------------------------------------------------------------------------
// d_out layout (flat, return order):
//   [0, 24896)            min_dists  (float)
//   [24896, 49792)        close_mask (0.0 / 1.0)
//   [49792, 51392)        contact_indices (int32 bit pattern)
//   [51392, 52992)        contact_dists (float, ascending)
// ---------------------------------------------------------------------------
extern "C" void kernel_launch(void* const* d_in, const int* in_sizes, int n_in,
                              void* d_out, int out_size, void* d_ws, size_t ws_size,
                              hipStream_t stream) {
    const float* hand = (const float*)d_in[0];   // [32, 778, 3]
    const float* obj  = (const float*)d_in[1];   // [32, 8192, 3]
    // d_in[2] = iteration (unused by the computation)

    float* out_min  = (float*)d_out;
    float* out_mask = out_min + BATCH * NH;
    int*   out_idx  = (int*)(out_mask + BATCH * NH);
    float* out_dist = (float*)(out_idx + BATCH * K_CONTACT);

    dim3 grid1(BATCH, HTILES);
    acz_min_dist_kernel<<<grid1, WAVES * 32, 0, stream>>>(hand, obj, out_min, out_mask);

    acz_topk_kernel<<<BATCH, 256, 0, stream>>>(out_min, out_idx, out_dist);
}